// GraphLayerAtt_20263655702718
// MI455X (gfx1250) — compile-verified
//
#include <hip/hip_runtime.h>
#include <hip/hip_bf16.h>

// GraphLayerAtt for MI455X (gfx1250, wave32, WMMA).
// Memory-bound (~1 GB traffic vs 33 GFLOP) -> fuse all edge GEMMs into one
// pass over x_a; keep fp32 via V_WMMA_F32_16X16X4_F32. Contiguous tile
// staging uses CDNA5 async global->LDS copies (ASYNCcnt) when available.

#define N_NODES 50000
#define N_EDGES 1000000
#define DIM 64

typedef float v2f __attribute__((ext_vector_type(2)));
typedef float v8f __attribute__((ext_vector_type(8)));

// ---- WMMA helpers (f32 16x16x4, wave32) -----------------------------------
// A 16x4 f32: lane L<16 holds A[M=L][k0], A[M=L][k0+1]; lane L>=16 holds
// A[M=L-16][k0+2], A[M=L-16][k0+3].
// B 4x16 f32 (B[k][n] = W[n][k] for y = x @ W.T): lane L<16 holds
// B[k0][n=L], B[k0+1][n]; lane L>=16 holds B[k0+2][n=L-16], B[k0+3][n].
// C/D 16x16 f32: vgpr v, lanes 0-15 -> (M=v, N=lane); lanes 16-31 -> (M=v+8).

__device__ __forceinline__ v8f wmma4(v2f a, v2f b, v8f c) {
    return __builtin_amdgcn_wmma_f32_16x16x4_f32(
        /*neg_a=*/false, a, /*neg_b=*/false, b,
        /*c_mod=*/(short)0, c, /*reuse_a=*/false, /*reuse_b=*/false);
}

__device__ __forceinline__ v2f lds_afrag(const float* tile /*[16][64]*/, int k0) {
    int lane = threadIdx.x & 31;
    int m = lane & 15;
    int k = k0 + ((lane >> 4) << 1);
    return *(const v2f*)(tile + m * DIM + k);
}

__device__ __forceinline__ v2f glb_bfrag(const float* __restrict__ W /*[64][64]*/,
                                         int j, int k0) {
    int lane = threadIdx.x & 31;
    int n = 16 * j + (lane & 15);
    int k = k0 + ((lane >> 4) << 1);     // even -> 8B aligned
    float2 t = *(const float2*)(W + n * DIM + k);
    v2f b; b.x = t.x; b.y = t.y;
    return b;
}

// ---- CDNA5 async global->LDS staging (ASYNCcnt path) ----------------------
#if __has_builtin(__builtin_amdgcn_global_load_async_to_lds_b128)
#define HAS_ASYNC_LDS 1
#else
#define HAS_ASYNC_LDS 0
#endif

#if HAS_ASYNC_LDS
typedef int vint4 __attribute__((vector_size(16)));
typedef __attribute__((address_space(1))) vint4 g_vint4;   // global int4
typedef __attribute__((address_space(3))) vint4 l_vint4;   // LDS int4
#endif

// Stage one contiguous 16x64 f32 tile (4 KB) into LDS; 8 x b128 per lane.
__device__ __forceinline__ void stage_tile(const float* g, float* ldsTile, int lane) {
#if HAS_ASYNC_LDS
    g_vint4* gp = (g_vint4*)(g);
    l_vint4* lp = (l_vint4*)(ldsTile);
#pragma unroll
    for (int i = 0; i < 8; ++i)
        __builtin_amdgcn_global_load_async_to_lds_b128(
            gp + lane + 32 * i, lp + lane + 32 * i, /*offset=*/0, /*cpol=*/0);
#else
    const float4* s = (const float4*)g;
    float4* d = (float4*)ldsTile;
#pragma unroll
    for (int i = 0; i < 8; ++i) d[lane + 32 * i] = s[lane + 32 * i];
#endif
}

__device__ __forceinline__ void wait_stage() {
#if HAS_ASYNC_LDS
#if __has_builtin(__builtin_amdgcn_s_wait_asynccnt)
    __builtin_amdgcn_s_wait_asynccnt(0);
#else
    asm volatile("s_wait_asynccnt 0" ::: "memory");
#endif
#endif
}

// Sign-split float atomic max (works for mixed signs, init = -inf).
__device__ __forceinline__ void atomicMaxF(float* addr, float v) {
    if (v >= 0.0f) atomicMax((int*)addr, __float_as_int(v));
    else           atomicMin((unsigned int*)addr, __float_as_uint(v));
}

// ---- Kernel 0: init segment-max / denom scratch ---------------------------
__global__ void __launch_bounds__(256) init_seg(float* maxb, float* denom) {
    int i = blockIdx.x * blockDim.x + threadIdx.x;
    if (i < N_NODES) { maxb[i] = -__builtin_inff(); denom[i] = 0.0f; }
}

// ---- Kernel 1: h_self = x_v @ W_vv.T + b_vv (wave per 16-row tile) --------
__global__ void __launch_bounds__(32) node_gemm(const float* __restrict__ xv,
                                                const float* __restrict__ Wvv,
                                                const float* __restrict__ bvv,
                                                float* __restrict__ hv) {
    __shared__ float tile[16 * DIM];
    const int base = blockIdx.x * 16;
    const int lane = threadIdx.x;

    stage_tile(xv + (size_t)base * DIM, tile, lane);
    wait_stage();
    __syncthreads();

    const int nloc = lane & 15;
    const int mbase = (lane >> 4) << 3;
#pragma unroll
    for (int j = 0; j < 4; ++j) {
        v8f acc = {};
#pragma unroll
        for (int k0 = 0; k0 < DIM; k0 += 4)
            acc = wmma4(lds_afrag(tile, k0), glb_bfrag(Wvv, j, k0), acc);
        int n = 16 * j + nloc;
        float bias = bvv[n];
#pragma unroll
        for (int v = 0; v < 8; ++v)
            hv[(size_t)(base + mbase + v) * DIM + n] = acc[v] + bias;
    }
}

// ---- Kernel 2: e = x_a @ W_att.T + b_att; segment max (wave per edge) -----
__global__ void __launch_bounds__(256) edge_logits(const float* __restrict__ xa,
                                                   const float* __restrict__ Watt,
                                                   const float* __restrict__ batt,
                                                   const int* __restrict__ dst,
                                                   float* __restrict__ eb,
                                                   float* __restrict__ maxb) {
    int gid = blockIdx.x * blockDim.x + threadIdx.x;
    int e = gid >> 5;                 // one wave32 per edge
    int lane = threadIdx.x & 31;
    if (e >= N_EDGES) return;
    const float* row = xa + (size_t)e * DIM;
    float s = row[lane] * Watt[lane] + row[lane + 32] * Watt[lane + 32];
#pragma unroll
    for (int off = 16; off; off >>= 1) s += __shfl_xor(s, off, 32);
    if (lane == 0) {
        float lg = s + batt[0];
        eb[e] = lg;
        atomicMaxF(&maxb[dst[e]], lg);
    }
}

// ---- Kernel 3: exp_e = exp(e - max[dst]); denom += exp_e ------------------
__global__ void __launch_bounds__(256) edge_exp(const int* __restrict__ dst,
                                                const float* __restrict__ maxb,
                                                float* __restrict__ eb,
                                                float* __restrict__ denom) {
    int i = blockIdx.x * blockDim.x + threadIdx.x;
    if (i >= N_EDGES) return;
    int d = dst[i];
    float ex = __expf(eb[i] - maxb[d]);
    eb[i] = ex;
    atomicAdd(&denom[d], ex);
}

// ---- Kernel 4: fused edge pass (wave per 16-edge tile) --------------------
// h_a = x_a@W_aa.T + b_aa + (x_v[src]+x_v[dst])@W_av.T
// h_v += scatter_dst( (x_a@W_va.T) * attn )
__global__ void __launch_bounds__(32) edge_tile(const float* __restrict__ xa,
                                                const float* __restrict__ xv,
                                                const int* __restrict__ srcIdx,
                                                const int* __restrict__ dstIdx,
                                                const float* __restrict__ Waa,
                                                const float* __restrict__ baa,
                                                const float* __restrict__ Wav,
                                                const float* __restrict__ Wva,
                                                const float* __restrict__ eb,
                                                const float* __restrict__ denom,
                                                float* __restrict__ hv,
                                                float* __restrict__ ha) {
    __shared__ float tA[16 * DIM];   // x_a tile
    __shared__ float tV[16 * DIM];   // x_v[src] + x_v[dst] tile
    const int base = blockIdx.x * 16;
    const int lane = threadIdx.x;

    // Kick off async x_a tile stage first (overlaps with gather below).
    stage_tile(xa + (size_t)base * DIM, tA, lane);

    // Per-row (edge) metadata in lanes 0..15, broadcast later via shfl.
    int   myDst = 0;
    float myAttn = 0.0f;
    if (lane < 16) {
        int e = base + lane;
        myDst = dstIdx[e];
        myAttn = eb[e] / denom[myDst];
    }

    // Stage gathered x_v[src]+x_v[dst] tile: lane -> (row = lane/2, half).
    // Needs the add, so this path stays through VGPRs.
    {
        int r = lane >> 1, h = lane & 1;
        int e = base + r;
        const float4* ps = (const float4*)(xv + (size_t)srcIdx[e] * DIM) + h * 8;
        const float4* pd = (const float4*)(xv + (size_t)dstIdx[e] * DIM) + h * 8;
        float4* dv = (float4*)(tV + r * DIM) + h * 8;
#pragma unroll
        for (int i = 0; i < 8; ++i) {
            float4 a = ps[i], b = pd[i];
            float4 c; c.x = a.x + b.x; c.y = a.y + b.y;
                      c.z = a.z + b.z; c.w = a.w + b.w;
            dv[i] = c;
        }
    }
    wait_stage();
    __syncthreads();

    const int nloc = lane & 15;
    const int mbase = (lane >> 4) << 3;
#pragma unroll
    for (int j = 0; j < 4; ++j) {
        v8f accAA = {}, accAV = {}, accVA = {};
#pragma unroll
        for (int k0 = 0; k0 < DIM; k0 += 4) {
            v2f aXA = lds_afrag(tA, k0);
            v2f aVS = lds_afrag(tV, k0);
            accAA = wmma4(aXA, glb_bfrag(Waa, j, k0), accAA);
            accAV = wmma4(aVS, glb_bfrag(Wav, j, k0), accAV);
            accVA = wmma4(aXA, glb_bfrag(Wva, j, k0), accVA);
        }
        int n = 16 * j + nloc;
        float bias = baa[n];
#pragma unroll
        for (int v = 0; v < 8; ++v) {
            int m = mbase + v;
            ha[(size_t)(base + m) * DIM + n] = accAA[v] + accAV[v] + bias;
            float a_m = __shfl(myAttn, m, 32);
            int   d_m = __shfl(myDst, m, 32);
            atomicAdd(&hv[(size_t)d_m * DIM + n], accVA[v] * a_m);
        }
    }
}

// ---- Launch ----------------------------------------------------------------
extern "C" void kernel_launch(void* const* d_in, const int* in_sizes, int n_in,
                              void* d_out, int out_size, void* d_ws, size_t ws_size,
                              hipStream_t stream) {
    const float* x_v  = (const float*)d_in[0];
    const float* x_a  = (const float*)d_in[1];
    const int*   arc  = (const int*)  d_in[2];
    const float* W_vv = (const float*)d_in[3];
    const float* b_vv = (const float*)d_in[4];
    const float* W_va = (const float*)d_in[5];
    const float* W_att= (const float*)d_in[6];
    const float* b_att= (const float*)d_in[7];
    const float* W_aa = (const float*)d_in[8];
    const float* b_aa = (const float*)d_in[9];
    const float* W_av = (const float*)d_in[10];

    const int* srcIdx = arc;              // arc_index[0]
    const int* dstIdx = arc + N_EDGES;    // arc_index[1]

    float* hv = (float*)d_out;                              // [N_NODES, 64]
    float* ha = (float*)d_out + (size_t)N_NODES * DIM;      // [N_EDGES, 64]

    // Scratch: exp_e [E], seg-max [N], denom [N]  (~4.4 MB)
    float* eb    = (float*)d_ws;
    float* maxb  = eb + N_EDGES;
    float* denom = maxb + N_NODES;

    init_seg   <<<(N_NODES + 255) / 256, 256, 0, stream>>>(maxb, denom);
    node_gemm  <<<N_NODES / 16,          32,  0, stream>>>(x_v, W_vv, b_vv, hv);
    edge_logits<<<(N_EDGES * 32) / 256,  256, 0, stream>>>(x_a, W_att, b_att,
                                                           dstIdx, eb, maxb);
    edge_exp   <<<(N_EDGES + 255) / 256, 256, 0, stream>>>(dstIdx, maxb, eb, denom);
    edge_tile  <<<N_EDGES / 16,          32,  0, stream>>>(x_a, x_v, srcIdx, dstIdx,
                                                           W_aa, b_aa, W_av, W_va,
                                                           eb, denom, hv, ha);
}